// AttentionHead_83726092468861
// MI455X (gfx1250) — compile-verified
//
#include <hip/hip_runtime.h>
#include <math.h>

// CDNA5 / gfx1250, wave32. fp32 WMMA path: V_WMMA_F32_16X16X4_F32.
typedef float v2f __attribute__((ext_vector_type(2)));
typedef float v8f __attribute__((ext_vector_type(8)));
typedef int   v4i __attribute__((ext_vector_type(4)));

// Pointer types for the async global->LDS builtin:
// (addrspace(1) v4i* src, addrspace(3) v4i* dst, imm offset, imm cpol)
typedef __attribute__((address_space(1))) v4i* gbl_v4i_ptr;
typedef __attribute__((address_space(3))) v4i* lds_v4i_ptr;

#define WMMA_F32(A, B, C) \
  __builtin_amdgcn_wmma_f32_16x16x4_f32(false, (A), false, (B), (short)0, (C), false, false)

#if defined(__has_builtin)
#  if __has_builtin(__builtin_amdgcn_global_load_async_to_lds_b128) && \
      __has_builtin(__builtin_amdgcn_s_wait_asynccnt)
#    define USE_ASYNC_LDS 1
#  endif
#endif
#ifndef USE_ASYNC_LDS
#  define USE_ASYNC_LDS 0
#endif

constexpr int BSZ = 4;
constexpr int T   = 2048;
constexpr int DM  = 1024;   // d_model
constexpr int DI  = 64;     // d_internal

// ---------------------------------------------------------------------------
// Stage 1: Out = x @ W^T for one projection matrix (launched once for Q with
// W_Q, once for K with W_K; V == K per reference bug so no third pass).
// One wave per 16-row tile of x; 4 WMMA chains (4 col tiles of 16).
// Hand-modulo-scheduled (unroll-2, operand register sets A/B). Only 4
// accumulator chains (32 VGPRs) so the allocator can pin them in place —
// the fused Q+K version (8 chains) forced per-iteration accumulator copies.
// x is 32 MB and fits in the 192 MB L2, so the second pass re-reads from L2.
// ---------------------------------------------------------------------------
__global__ __launch_bounds__(32) void proj_one(const float* __restrict__ x,
                                               const float* __restrict__ W,
                                               float* __restrict__ Out)
{
  const int lane = threadIdx.x & 31;
  const int half = lane >> 4;
  const int l16  = lane & 15;
  const int kOff = half * 2;
  const int rt   = blockIdx.x;                       // 16-row tile (0..511)

  const float* xp = x + ((size_t)rt * 16 + l16) * DM + kOff;
  const float* wp = W + (size_t)l16 * DM + kOff;     // + nt*16*DM per col tile

  v8f acc[4] = {};

  // Two operand register sets for software pipelining (no rotation copies).
  v2f aA, bA[4];
  v2f aB, bB[4];

  auto loadA = [&](int k) {
    aA = *(const v2f*)(xp + k);
#pragma unroll
    for (int nt = 0; nt < 4; ++nt)
      bA[nt] = *(const v2f*)(wp + (size_t)nt * 16 * DM + k);
  };
  auto loadB = [&](int k) {
    aB = *(const v2f*)(xp + k);
#pragma unroll
    for (int nt = 0; nt < 4; ++nt)
      bB[nt] = *(const v2f*)(wp + (size_t)nt * 16 * DM + k);
  };
  auto computeA = [&]() {
#pragma unroll
    for (int nt = 0; nt < 4; ++nt) acc[nt] = WMMA_F32(aA, bA[nt], acc[nt]);
  };
  auto computeB = [&]() {
#pragma unroll
    for (int nt = 0; nt < 4; ++nt) acc[nt] = WMMA_F32(aB, bB[nt], acc[nt]);
  };

  loadA(0);
  for (int k = 0; k < DM - 8; k += 8) {
    loadB(k + 4);   // prefetch while computing on set A
    computeA();
    loadA(k + 8);   // prefetch while computing on set B
    computeB();
  }
  loadB(DM - 4);    // last k-step
  computeA();
  computeB();

  // C layout: reg r, lane -> row = r + 8*half, col = nt*16 + l16
#pragma unroll
  for (int nt = 0; nt < 4; ++nt) {
#pragma unroll
    for (int r = 0; r < 8; ++r) {
      size_t orow = (size_t)rt * 16 + half * 8 + r;
      Out[orow * DI + nt * 16 + l16] = acc[nt][r];
    }
  }
}

// ---------------------------------------------------------------------------
// Stage 2: fused causal flash attention, V = K.
// One wave per (batch, 16-query tile). K tile double-buffered in LDS with
// async global->LDS copies (ASYNCcnt); online softmax with cross-lane
// reductions; P transposed C->A layout through LDS.
// ---------------------------------------------------------------------------
__global__ __launch_bounds__(32) void attn(const float* __restrict__ Q,
                                           const float* __restrict__ Kg,
                                           float* __restrict__ out)
{
  __shared__ float sK[2][16 * DI];  // 2 x 4 KB: double-buffered K/V tile
  __shared__ float sP[16 * 16];     // 1 KB: prob tile for layout transpose

  const int lane = threadIdx.x & 31;
  const int half = lane >> 4;
  const int l16  = lane & 15;
  const int kOff = half * 2;
  const int qt   = blockIdx.x;
  const int b    = blockIdx.y;
  const size_t qbase = (size_t)b * T + (size_t)qt * 16;

  // Q tile in A-layout registers: row M = l16, k-pair at 4*j + kOff
  v2f qa[16];
  {
    const float* qrow = Q + (qbase + l16) * DI + kOff;
#pragma unroll
    for (int j = 0; j < 16; ++j) qa[j] = *(const v2f*)(qrow + 4 * j);
  }

  v8f o[4] = {};
  float m[8], l[8];
#pragma unroll
  for (int r = 0; r < 8; ++r) { m[r] = -INFINITY; l[r] = 0.0f; }

  const float scale = 0.03125f;  // 1/sqrt(d_model)

  // ---- tile loader: global K tile (16x64 fp32) -> LDS buffer ----
  auto load_tile = [&](int st, int buf) {
    const float* kt = Kg + ((size_t)b * T + (size_t)st * 16) * DI;
#if USE_ASYNC_LDS
#pragma unroll
    for (int i = 0; i < 8; ++i) {
      const float* g = kt + (size_t)(i * 32 + lane) * 4;
      __builtin_amdgcn_global_load_async_to_lds_b128(
          (gbl_v4i_ptr)(g),
          (lds_v4i_ptr)(&sK[buf][(i * 32 + lane) * 4]),
          0, 0);
    }
#else
#pragma unroll
    for (int i = 0; i < 8; ++i) {
      float4 v = *(const float4*)(kt + (size_t)(i * 32 + lane) * 4);
      *(float4*)(&sK[buf][(i * 32 + lane) * 4]) = v;
    }
#endif
  };

  load_tile(0, 0);

  for (int st = 0; st <= qt; ++st) {
    const int cur = st & 1;
    const bool pf = (st < qt);
    if (pf) load_tile(st + 1, cur ^ 1);   // prefetch next tile into other buffer

#if USE_ASYNC_LDS
    if (pf) __builtin_amdgcn_s_wait_asynccnt(8);  // current tile done, next in flight
    else    __builtin_amdgcn_s_wait_asynccnt(0);
    asm volatile("" ::: "memory");                 // keep LDS reads below the wait
#else
    __syncthreads();
#endif
    const float* sKc = sK[cur];

    // ---- S = Q K^T : batch all 16 LDS operand loads, then the WMMA chain ----
    v2f kb[16];
#pragma unroll
    for (int j = 0; j < 16; ++j)
      kb[j] = *(const v2f*)(&sKc[l16 * DI + 4 * j + kOff]);
    v8f S = {};
#pragma unroll
    for (int j = 0; j < 16; ++j) S = WMMA_F32(qa[j], kb[j], S);

    // ---- scale, causal mask, online softmax update ----
    const int kglob = st * 16 + l16;       // key column handled by this lane
    const int qrow0 = qt * 16 + half * 8;  // query row of C-register 0
    float p[8], alpha[8];
#pragma unroll
    for (int r = 0; r < 8; ++r) {
      float s = S[r] * scale;
      if (kglob > qrow0 + r) s = -INFINITY;
      float t = s;
      t = fmaxf(t, __shfl_xor(t, 1, 32));
      t = fmaxf(t, __shfl_xor(t, 2, 32));
      t = fmaxf(t, __shfl_xor(t, 4, 32));
      t = fmaxf(t, __shfl_xor(t, 8, 32));
      float mnew = fmaxf(m[r], t);
      alpha[r] = __expf(m[r] - mnew);
      m[r] = mnew;
      float pe = __expf(s - mnew);
      p[r] = pe;
      float su = pe;
      su += __shfl_xor(su, 1, 32);
      su += __shfl_xor(su, 2, 32);
      su += __shfl_xor(su, 4, 32);
      su += __shfl_xor(su, 8, 32);
      l[r] = l[r] * alpha[r] + su;
    }
#pragma unroll
    for (int nt = 0; nt < 4; ++nt)
#pragma unroll
      for (int r = 0; r < 8; ++r) o[nt][r] *= alpha[r];

    // ---- transpose P: C layout -> A layout via LDS ----
    __syncthreads();
#pragma unroll
    for (int r = 0; r < 8; ++r) sP[(r + half * 8) * 16 + l16] = p[r];
    __syncthreads();
    v2f pa[4];
#pragma unroll
    for (int j = 0; j < 4; ++j)
      pa[j] = *(const v2f*)(&sP[l16 * 16 + 4 * j + kOff]);

    // ---- O += P @ V (V = K tile): batch 16 operand loads, then 16 WMMAs ----
    v2f bv[16];
#pragma unroll
    for (int j = 0; j < 4; ++j) {
#pragma unroll
      for (int nt = 0; nt < 4; ++nt) {
        bv[j * 4 + nt].x = sKc[(4 * j + kOff + 0) * DI + nt * 16 + l16];
        bv[j * 4 + nt].y = sKc[(4 * j + kOff + 1) * DI + nt * 16 + l16];
      }
    }
#pragma unroll
    for (int j = 0; j < 4; ++j)
#pragma unroll
      for (int nt = 0; nt < 4; ++nt)
        o[nt] = WMMA_F32(pa[j], bv[j * 4 + nt], o[nt]);

#if !USE_ASYNC_LDS
    __syncthreads();   // before next iteration's prefetch overwrites this buffer
#endif
  }

  // ---- normalize and store ----
#pragma unroll
  for (int r = 0; r < 8; ++r) {
    float inv = 1.0f / l[r];
    size_t orow = (qbase + half * 8 + r) * DI;
#pragma unroll
    for (int nt = 0; nt < 4; ++nt)
      out[orow + nt * 16 + l16] = o[nt][r] * inv;
  }
}

// ---------------------------------------------------------------------------
extern "C" void kernel_launch(void* const* d_in, const int* in_sizes, int n_in,
                              void* d_out, int out_size, void* d_ws, size_t ws_size,
                              hipStream_t stream) {
  const float* x  = (const float*)d_in[0];
  const float* Wq = (const float*)d_in[1];
  const float* Wk = (const float*)d_in[2];
  // d_in[3] (W_V) intentionally unused: reference computes V with W_K.
  float* out = (float*)d_out;

  float* Qbuf = (float*)d_ws;                       // [B*T, 64] = 2 MB
  float* Kbuf = Qbuf + (size_t)BSZ * T * DI;        // [B*T, 64] = 2 MB

  proj_one<<<dim3(BSZ * T / 16), 32, 0, stream>>>(x, Wq, Qbuf);
  proj_one<<<dim3(BSZ * T / 16), 32, 0, stream>>>(x, Wk, Kbuf);
  attn<<<dim3(T / 16, BSZ), 32, 0, stream>>>(Qbuf, Kbuf, out);
}